// LinearFullRNN_77232101917082
// MI455X (gfx1250) — compile-verified
//
#include <hip/hip_runtime.h>

// theta_{t+1} = A @ theta_t + B @ (theta_t - theta_{t-1}), 50 steps, d=4096, f32.
// Memory-bound (0.5 FLOP/byte). A+B = 134 MB fits MI455X's 192 MB L2, so after
// the first step every launch is L2-resident. One fused step kernel, 50
// stream-ordered launches; dot products run on V_WMMA_F32_16X16X4_F32 (full f32
// precision), matrix tile as the WMMA A-operand, theta/delta broadcast into the
// B-operand columns for free (per-lane address depends only on lane>>4).

typedef float v2f __attribute__((ext_vector_type(2)));
typedef float v4f __attribute__((ext_vector_type(4)));
typedef float v8f __attribute__((ext_vector_type(8)));

#define DIM 4096
#define STEPS 50
#define WAVES_PER_WG 4
#define K_PER_WAVE (DIM / WAVES_PER_WG) // 1024

__global__ __launch_bounds__(32 * WAVES_PER_WG) void
rnn_step_wmma(const float* __restrict__ A,
              const float* __restrict__ Bm,
              const float* __restrict__ theta,
              const float* __restrict__ theta_prev,
              float* __restrict__ out)
{
    __shared__ float red[16];

    const int lane = threadIdx.x & 31;
    const int wave = threadIdx.x >> 5;
    const int h    = lane >> 4;   // which half-wave: selects K-slots {0,1} vs {2,3}
    const int m    = lane & 15;   // row within the 16-row block
    const int row  = blockIdx.x * 16 + m;

    if (threadIdx.x < 16) red[threadIdx.x] = 0.0f;
    __syncthreads();

    const float* __restrict__ arow = A  + (size_t)row * DIM;
    const float* __restrict__ brow = Bm + (size_t)row * DIM;

    v8f acc = {};
    const int k0 = wave * K_PER_WAVE;
    const int k1 = k0 + K_PER_WAVE;

    // Per iteration: K=8 columns via two 16x16x4 f32 WMMAs per matrix.
    // Lane loads a float4 at column (k + 4*h): elements 0,1 feed WMMA slots
    // K={0,1} (low lanes) / K={2,3} (high lanes) of tile0; elements 2,3 feed
    // tile1. The vector operand uses the identical addressing, so the A-tile
    // column set and the x/delta values line up slot-for-slot.
    #pragma unroll 4
    for (int k = k0; k < k1; k += 8) {
        const int off = k + 4 * h;
        v4f a4 = *(const v4f*)(arow + off);
        v4f b4 = *(const v4f*)(brow + off);
        v4f x4 = *(const v4f*)(theta + off);
        v4f p4 = *(const v4f*)(theta_prev + off);
        v4f d4 = x4 - p4;

        v2f aA0 = {a4.x, a4.y}, aA1 = {a4.z, a4.w};
        v2f bA0 = {b4.x, b4.y}, bA1 = {b4.z, b4.w};
        v2f xB0 = {x4.x, x4.y}, xB1 = {x4.z, x4.w};
        v2f dB0 = {d4.x, d4.y}, dB1 = {d4.z, d4.w};

        acc = __builtin_amdgcn_wmma_f32_16x16x4_f32(false, aA0, false, xB0,
                                                    (short)0, acc, false, false);
        acc = __builtin_amdgcn_wmma_f32_16x16x4_f32(false, aA1, false, xB1,
                                                    (short)0, acc, false, false);
        acc = __builtin_amdgcn_wmma_f32_16x16x4_f32(false, bA0, false, dB0,
                                                    (short)0, acc, false, false);
        acc = __builtin_amdgcn_wmma_f32_16x16x4_f32(false, bA1, false, dB1,
                                                    (short)0, acc, false, false);
    }

    // D layout: VGPR v, lane l -> M = v + 8*(l>>4), N = l&15. Every column N
    // holds the same y (B columns were broadcast), so lanes with N==0 (lanes 0
    // and 16) carry the 16 partial results; reduce the 4 K-split waves in LDS.
    if (m == 0) {
        #pragma unroll
        for (int v = 0; v < 8; ++v)
            atomicAdd(&red[v + 8 * h], acc[v]); // ds_add_f32
    }
    __syncthreads();
    if (threadIdx.x < 16)
        out[blockIdx.x * 16 + threadIdx.x] = red[threadIdx.x];
}

extern "C" void kernel_launch(void* const* d_in, const int* in_sizes, int n_in,
                              void* d_out, int out_size, void* d_ws, size_t ws_size,
                              hipStream_t stream)
{
    (void)in_sizes; (void)n_in; (void)d_ws; (void)ws_size; (void)out_size;

    const float* init_seq = (const float*)d_in[0]; // [2, DIM]
    const float* A        = (const float*)d_in[1]; // [DIM, DIM]
    const float* B        = (const float*)d_in[2]; // [DIM, DIM]
    float*       out      = (float*)d_out;         // [STEPS, DIM]

    const float* th_last  = init_seq + DIM; // init_seq[-1] = theta_0
    const float* th_last2 = init_seq;       // init_seq[-2] = theta_{-1}

    dim3 grid(DIM / 16);
    dim3 block(32 * WAVES_PER_WG);

    for (int t = 0; t < STEPS; ++t) {
        const float* theta = (t == 0) ? th_last
                                      : out + (size_t)(t - 1) * DIM;
        const float* prev  = (t == 0) ? th_last2
                           : (t == 1) ? th_last
                                      : out + (size_t)(t - 2) * DIM;
        rnn_step_wmma<<<grid, block, 0, stream>>>(A, B, theta, prev,
                                                  out + (size_t)t * DIM);
    }
}